// MultiPlaneSliceProject_17025250362017
// MI455X (gfx1250) — compile-verified
//
#include <hip/hip_runtime.h>
#include <hip/hip_bf16.h>

// ---------------------------------------------------------------------------
// MultiPlaneSliceProject for MI455X (gfx1250, wave32)
//
// Stage 1 (HBM-bound, ~23us @ 23.3TB/s): square+triple-mean 512MiB of x into
//          feat [B,192,C] (B=4, C=128, N=64).
// Stage 2: feat[768,128] @ W1[128,1536] + b1, ReLU  -> h[768,1536]   (WMMA f32)
// Stage 3: h[768,1536]   @ W2[1536,768] + b2        -> out[768,768]  (WMMA f32)
// ---------------------------------------------------------------------------

typedef __attribute__((ext_vector_type(2))) float v2f;
typedef __attribute__((ext_vector_type(8))) float v8f;

#define BDIM 4
#define CDIM 128
#define NDIM 64
#define HID 1536
#define PROJ 768
#define ROWS 192              // 3*NDIM positions per batch
#define MTOT (BDIM * ROWS)    // 768 GEMM rows

// ---------------------------------------------------------------------------
// Stage 1: per-(b,c) cube reduction.
// grid = B*C = 512 blocks, block = 256 threads (8 waves).
// Each wave handles rows (d,h) = wave, wave+8, ... (4096 rows of 64 w-floats).
// lane loads float2 at w = 2*lane -> fully coalesced 256B/wave/issue.
// ---------------------------------------------------------------------------
__global__ __launch_bounds__(256) void reduce_kernel(
    const float* __restrict__ x, float* __restrict__ feat) {
  const int bc = blockIdx.x;            // 0..511
  const int b  = bc >> 7;               // / C
  const int c  = bc & (CDIM - 1);
  const float* __restrict__ base = x + (size_t)bc * (NDIM * NDIM * NDIM);

  __shared__ float sD[NDIM];
  __shared__ float sH[NDIM];
  __shared__ float sW[NDIM];

  const int t = threadIdx.x;
  if (t < NDIM) { sD[t] = 0.0f; sH[t] = 0.0f; sW[t] = 0.0f; }
  __syncthreads();

  const int wave = t >> 5;              // 0..7
  const int lane = t & 31;
  const int w0 = lane << 1;             // w = 2*lane, 2*lane+1

  float accW0 = 0.0f, accW1 = 0.0f;

  #pragma unroll 4
  for (int r = wave; r < NDIM * NDIM; r += 8) {   // r = d*64 + h
    const v2f v = *(const v2f*)(base + (size_t)r * NDIM + w0);
    const float s0 = v.x * v.x;
    const float s1 = v.y * v.y;
    accW0 += s0;
    accW1 += s1;
    float rowSum = s0 + s1;
    #pragma unroll
    for (int off = 16; off > 0; off >>= 1)
      rowSum += __shfl_xor(rowSum, off, 32);
    if (lane == 0) {
      atomicAdd(&sD[r >> 6], rowSum);           // sum over (h,w) for this d
      atomicAdd(&sH[r & (NDIM - 1)], rowSum);   // sum over (d,w) for this h
    }
  }
  atomicAdd(&sW[w0], accW0);
  atomicAdd(&sW[w0 + 1], accW1);
  __syncthreads();

  // feat[b, pos, c]; pos 0..63 = sc(D), 64..127 = ss(H), 128..191 = sa(W)
  if (t < ROWS) {
    const float v = (t < 64) ? sD[t] : (t < 128) ? sH[t - 64] : sW[t - 128];
    feat[((size_t)b * ROWS + t) * CDIM + c] = v * (1.0f / (NDIM * NDIM));
  }
}

// ---------------------------------------------------------------------------
// WMMA f32 GEMM: D = [relu](A[M,K] @ Bm[K,N] + bias[N]), all row-major f32.
// One wave per 16x16 tile; K-loop over V_WMMA_F32_16X16X4_F32.
//
// Fragment layouts (ISA 7.12.2, 32-bit, wave32):
//   A 16x4 : lanes 0-15 hold M=lane, {VGPR0,VGPR1} = K{0,1}; lanes 16-31 K{2,3}
//   B 4x16 : lanes 0-15 hold N=lane, {VGPR0,VGPR1} = K{0,1}; lanes 16-31 K{2,3}
//   C/D    : VGPR r = row (r + 8*half), N = lane&15
// ---------------------------------------------------------------------------
template <bool RELU>
__global__ __launch_bounds__(256) void gemm_wmma(
    const float* __restrict__ A, const float* __restrict__ Bm,
    const float* __restrict__ bias, float* __restrict__ D,
    int M, int K, int N) {
  const int lane    = threadIdx.x & 31;
  const int waveId  = blockIdx.x * (blockDim.x >> 5) + (threadIdx.x >> 5);
  const int tilesN  = N >> 4;
  const int tileM   = waveId / tilesN;
  const int tileN   = waveId - tileM * tilesN;
  if ((tileM << 4) >= M) return;        // wave-uniform: EXEC stays all-ones

  const int half = lane >> 4;           // 0 => K{0,1}, 1 => K{2,3}
  const int l16  = lane & 15;

  const float* __restrict__ aRow  = A + (size_t)((tileM << 4) + l16) * K + (half << 1);
  const float* __restrict__ bBase = Bm + (size_t)(half << 1) * N + (tileN << 4) + l16;

  v8f acc = {};
  for (int k = 0; k < K; k += 4) {
    const v2f a = *(const v2f*)(aRow + k);        // A[m][k+2h], A[m][k+2h+1]
    v2f bfrag;
    bfrag.x = bBase[(size_t)k * N];               // B[k+2h  ][n]
    bfrag.y = bBase[(size_t)(k + 1) * N];         // B[k+2h+1][n]
    acc = __builtin_amdgcn_wmma_f32_16x16x4_f32(
        /*neg_a=*/false, a, /*neg_b=*/false, bfrag,
        /*c_mod=*/(short)0, acc, /*reuse_a=*/false, /*reuse_b=*/false);
  }

  const float bv = bias[(tileN << 4) + l16];
  float* __restrict__ dBase =
      D + (size_t)((tileM << 4) + (half << 3)) * N + (tileN << 4) + l16;
  #pragma unroll
  for (int r = 0; r < 8; ++r) {
    float v = acc[r] + bv;
    if (RELU) v = fmaxf(v, 0.0f);
    dBase[(size_t)r * N] = v;
  }
}

// ---------------------------------------------------------------------------
extern "C" void kernel_launch(void* const* d_in, const int* in_sizes, int n_in,
                              void* d_out, int out_size, void* d_ws, size_t ws_size,
                              hipStream_t stream) {
  const float* x  = (const float*)d_in[0];   // [4,128,64,64,64]
  const float* W1 = (const float*)d_in[1];   // [128,1536]
  const float* b1 = (const float*)d_in[2];   // [1536]
  const float* W2 = (const float*)d_in[3];   // [1536,768]
  const float* b2 = (const float*)d_in[4];   // [768]
  float* out = (float*)d_out;                // [4,192,768]

  float* feat = (float*)d_ws;                // [768,128]  (384 KiB)
  float* h    = feat + (size_t)MTOT * CDIM;  // [768,1536] (4.5 MiB)

  // Stage 1: 512 cubes, one block each.
  reduce_kernel<<<BDIM * CDIM, 256, 0, stream>>>(x, feat);

  // Stage 2: 48x96 = 4608 tiles, 8 waves/block -> 576 blocks.
  {
    const int tiles = (MTOT / 16) * (HID / 16);
    gemm_wmma<true><<<tiles / 8, 256, 0, stream>>>(feat, W1, b1, h,
                                                   MTOT, CDIM, HID);
  }
  // Stage 3: 48x48 = 2304 tiles -> 288 blocks.
  {
    const int tiles = (MTOT / 16) * (PROJ / 16);
    gemm_wmma<false><<<tiles / 8, 256, 0, stream>>>(h, W2, b2, out,
                                                    MTOT, HID, PROJ);
  }
}